// PlanPredCollisionLoss_16844861735211
// MI455X (gfx1250) — compile-verified
//
#include <hip/hip_runtime.h>
#include <hip/hip_bf16.h>
#include <stdint.h>

#define T_STEPS 16
#define M_MODES 6
#define APB     512   // agents per block
#define BLK     256   // 8 waves (wave32)
#define NWAVES  (BLK / 32)
#define INF_BITS 0x7F800000u

// ---- CDNA5 async global->LDS availability probes -------------------------
#if defined(__has_builtin)
#  if __has_builtin(__builtin_amdgcn_global_load_async_to_lds_b128)
#    define HAVE_ASYNC_LDS 1
#  endif
#  if __has_builtin(__builtin_amdgcn_s_wait_asynccnt)
#    define HAVE_WAIT_ASYNC 1
#  endif
#endif
#ifndef HAVE_ASYNC_LDS
#define HAVE_ASYNC_LDS 0
#endif
#ifndef HAVE_WAIT_ASYNC
#define HAVE_WAIT_ASYNC 0
#endif

// Builtin params: pointers to 16-byte int vectors in AS(1) (global) / AS(3) (LDS).
typedef int v4i_t __attribute__((__vector_size__(16)));
typedef __attribute__((address_space(1))) v4i_t g_v4i_t;
typedef __attribute__((address_space(3))) v4i_t l_v4i_t;

__device__ __forceinline__ void async_copy_b128(const void* gsrc, void* ldst) {
#if HAVE_ASYNC_LDS
  __builtin_amdgcn_global_load_async_to_lds_b128(
      (g_v4i_t*)(uintptr_t)gsrc,
      (l_v4i_t*)(uint32_t)(uintptr_t)ldst,
      /*offset=*/0, /*cpol=*/0);
#else
  (void)gsrc; (void)ldst;
#endif
}

__device__ __forceinline__ void wait_asynccnt0() {
#if HAVE_WAIT_ASYNC
  __builtin_amdgcn_s_wait_asynccnt(0);
#else
  asm volatile("s_wait_asynccnt 0" ::: "memory");
#endif
}

// ---- kernel 0: init global min scratch to +inf bits ----------------------
__global__ void pp_init_ws(unsigned* __restrict__ wsmin, float* __restrict__ out, int n) {
  int i = blockIdx.x * blockDim.x + threadIdx.x;
  if (i < n) wsmin[i] = INF_BITS;   // +inf bits; uint min == float min for vals >= 0
  if (i == 0) out[0] = 0.0f;
}

// ---- kernel 1: lane==timestep layout, register min accumulation ----------
__global__ __launch_bounds__(BLK)
void pp_collide_min(const float* __restrict__ ego,     // [B,T,2]
                    const float* __restrict__ preds,   // [B,A,M,T,2]
                    const float* __restrict__ scores,  // [B,A,M]
                    unsigned*    __restrict__ wsmin,   // [2,B,T] (x mins, then y mins)
                    int A)
{
  const int b    = blockIdx.x;
  const int aBeg = blockIdx.y * APB;
  const int tid  = threadIdx.x;
  const int lane = tid & 31;
  const int wave = tid >> 5;
  const int BT   = gridDim.x * T_STEPS;      // offset to y-min region

  __shared__ __align__(16) float sScore[APB * M_MODES]; // 12 KB score slab
  __shared__ unsigned sOff[APB];  // per-agent float-index of best-mode tile
  __shared__ float    sLow[APB];  // per-agent penalty floor: 0 / 100 / +inf
  __shared__ unsigned sMinX[T_STEPS];
  __shared__ unsigned sMinY[T_STEPS];

  if (tid < T_STEPS) { sMinX[tid] = INF_BITS; sMinY[tid] = INF_BITS; }

  // Stage this block's contiguous score slab into LDS via CDNA5 async copy.
  {
    const char* gbase = (const char*)(scores + ((size_t)b * A + aBeg) * M_MODES);
    char*       lbase = (char*)sScore;
    const int n16 = (APB * M_MODES * 4) / 16;   // 768 x 16B
#if HAVE_ASYNC_LDS
    for (int i = tid; i < n16; i += BLK)
      async_copy_b128(gbase + (size_t)i * 16, lbase + (size_t)i * 16);
    wait_asynccnt0();
#else
    const float4* g4 = (const float4*)gbase;
    float4*       l4 = (float4*)lbase;
    for (int i = tid; i < n16; i += BLK) l4[i] = g4[i];
#endif
  }
  __syncthreads();

  // Pre-pass: per-agent argmax over modes (first-max wins == jnp.argmax),
  // publish best-mode tile offset + penalty floor.
  for (int ai = tid; ai < APB; ai += BLK) {
    const int a = aBeg + ai;
    if (a < A) {
      const float* sc = &sScore[ai * M_MODES];
      float best = sc[0]; int mi = 0;
#pragma unroll
      for (int m = 1; m < M_MODES; ++m) { float v = sc[m]; if (v > best) { best = v; mi = m; } }
      sOff[ai] = ((unsigned)((unsigned)b * (unsigned)A + (unsigned)a) * M_MODES + (unsigned)mi)
                 * (unsigned)(T_STEPS * 2);
      sLow[ai] = (best < 0.5f) ? 100.0f : 0.0f;
    } else {
      sOff[ai] = 0u;
      sLow[ai] = __uint_as_float(INF_BITS);  // sentinel: never wins the min
    }
  }
  __syncthreads();

  // Main loop: lane -> timestep. Each 16-lane half processes one agent per
  // iteration; its float2 loads form one contiguous 128B tile (coalesced).
  const int   t  = lane & 15;
  const int   hf = lane >> 4;
  const float ex = ego[(size_t)b * (T_STEPS * 2) + 2 * t];
  const float ey = ego[(size_t)b * (T_STEPS * 2) + 2 * t + 1];

  float xmin = __uint_as_float(INF_BITS);
  float ymin = __uint_as_float(INF_BITS);

  const int perWave = APB / NWAVES;          // 64 agents per wave
  const int base    = wave * perWave;
  for (int i = 0; i < perWave; i += 2) {
    const int      ai     = base + i + hf;   // one agent per half-wave
    const unsigned off    = sOff[ai];        // LDS broadcast (uniform per half)
    const float    lowPen = sLow[ai];
    const float2   p      = *(const float2*)(preds + (size_t)off + 2 * t);
    const float dx = ex - p.x;
    const float dy = ey - p.y;
    float msk = (dx * dx + dy * dy > 9.0f) ? 100.0f : 0.0f;
    msk = fmaxf(msk, lowPen);                // low-confidence => 100 (or +inf sentinel)
    xmin = fminf(xmin, fabsf(dx) + msk);
    ymin = fminf(ymin, fabsf(dy) + msk);
  }

  // Merge the two half-waves (same t), then 16-lane conflict-free LDS min.
  xmin = fminf(xmin, __shfl_xor(xmin, 16, 32));
  ymin = fminf(ymin, __shfl_xor(ymin, 16, 32));
  if (lane < T_STEPS) {
    atomicMin(&sMinX[t], __float_as_uint(xmin));
    atomicMin(&sMinY[t], __float_as_uint(ymin));
  }
  __syncthreads();

  if (tid < T_STEPS) {
    atomicMin(&wsmin[b * T_STEPS + tid],      sMinX[tid]);
    atomicMin(&wsmin[BT + b * T_STEPS + tid], sMinY[tid]);
  }
}

// ---- kernel 2: deterministic hinge-loss + mean ---------------------------
__global__ __launch_bounds__(BLK)
void pp_finalize(const unsigned* __restrict__ wsmin,   // [2,B,T]
                 const float*    __restrict__ mask,    // [B,T]
                 float*          __restrict__ out,
                 int B)
{
  __shared__ float part[BLK];
  const int tid = threadIdx.x;
  const int n = B * T_STEPS;
  float acc = 0.0f;
  for (int i = tid; i < n; i += BLK) {
    float xmn = __uint_as_float(wsmin[i]);
    float ymn = __uint_as_float(wsmin[n + i]);
    float xl = (xmn <= 1.5f) ? (1.5f - xmn) : 0.0f;
    float yl = (ymn <= 3.0f) ? (3.0f - ymn) : 0.0f;
    acc += (xl + yl) * mask[i];
  }
  part[tid] = acc;
  __syncthreads();
  for (int off = BLK / 2; off > 0; off >>= 1) {
    if (tid < off) part[tid] += part[tid + off];
    __syncthreads();
  }
  if (tid == 0) out[0] = part[0] * (1.0f / (float)(B * T_STEPS * 2));  // LOSS_WEIGHT=1
}

extern "C" void kernel_launch(void* const* d_in, const int* in_sizes, int n_in,
                              void* d_out, int out_size, void* d_ws, size_t ws_size,
                              hipStream_t stream) {
  const float* ego    = (const float*)d_in[0];   // [B,T,2]
  const float* preds  = (const float*)d_in[1];   // [B,A,M,T,2]
  const float* scores = (const float*)d_in[2];   // [B,A,M]
  const float* mask   = (const float*)d_in[3];   // [B,T]
  float* out = (float*)d_out;

  const int B = in_sizes[0] / (T_STEPS * 2);
  const int A = (int)((long long)in_sizes[2] / ((long long)B * M_MODES));

  unsigned* wsmin = (unsigned*)d_ws;             // 2*B*T uints (16 KB for B=128)
  const int nMin = 2 * B * T_STEPS;

  pp_init_ws<<<(nMin + BLK - 1) / BLK, BLK, 0, stream>>>(wsmin, out, nMin);

  dim3 grid(B, (A + APB - 1) / APB);
  pp_collide_min<<<grid, BLK, 0, stream>>>(ego, preds, scores, wsmin, A);

  pp_finalize<<<1, BLK, 0, stream>>>(wsmin, mask, out, B);
}